// Repro_56057913147433
// MI455X (gfx1250) — compile-verified
//
#include <hip/hip_runtime.h>
#include <cstdint>
#include <cstddef>

// Problem constants (from the reference).
#define NN 2
#define CC 3
#define HH 345
#define WW 456
// LDS staging tile (per block): 64 x 40 x 3 channels x 4B = 30720 B of LDS.
#define TW 64
#define TH 40

typedef unsigned int u32x4 __attribute__((ext_vector_type(4)));
typedef int          i32x8 __attribute__((ext_vector_type(8)));
typedef int          i32x4 __attribute__((ext_vector_type(4)));

// Cubic convolution weights, a = -0.75 (matches PyTorch/JAX reference).
__device__ __forceinline__ float cub1(float x) {
    // ((a+2)x - (a+3))x^2 + 1 = (1.25x - 2.25)x^2 + 1
    return fmaf(fmaf(1.25f, x, -2.25f) * x, x, 1.0f);
}
__device__ __forceinline__ float cub2(float x) {
    // ((a*x - 5a)x + 8a)x - 4a = ((-0.75x + 3.75)x - 6)x + 3
    return fmaf(fmaf(fmaf(-0.75f, x, 3.75f), x, -6.0f), x, 3.0f);
}

__global__ __launch_bounds__(256) void grid_sample_bicubic_tdm(
    const float* __restrict__ img,   // [N,C,H,W]
    const float* __restrict__ mat,   // [N,2,3]
    float* __restrict__ out)         // [N,C,H,W]
{
    __shared__ float sh[CC * TH * TW];

    const int n  = blockIdx.z;
    const int w0 = blockIdx.x * 16;
    const int h0 = blockIdx.y * 16;

    // Per-n affine coefficients: uniform per block -> scalar loads.
    const float m00 = mat[n * 6 + 0], m01 = mat[n * 6 + 1], m02 = mat[n * 6 + 2];
    const float m10 = mat[n * 6 + 3], m11 = mat[n * 6 + 4], m12 = mat[n * 6 + 5];

    // Sample coordinate in input pixel space for output pixel (w,h).
    auto sampx = [&](float w, float h) {
        float xs = fmaf(2.0f / WW, w, 1.0f / WW - 1.0f);
        float ys = fmaf(2.0f / HH, h, 1.0f / HH - 1.0f);
        float gx = fmaf(m00, xs, fmaf(m01, ys, m02));
        return fmaf(gx, WW * 0.5f, (WW - 1) * 0.5f);
    };
    auto sampy = [&](float w, float h) {
        float xs = fmaf(2.0f / WW, w, 1.0f / WW - 1.0f);
        float ys = fmaf(2.0f / HH, h, 1.0f / HH - 1.0f);
        float gy = fmaf(m10, xs, fmaf(m11, ys, m12));
        return fmaf(gy, HH * 0.5f, (HH - 1) * 0.5f);
    };

    // Affine => extrema of ix/iy over the block occur at the 4 block corners.
    float ax0 = sampx((float)w0,        (float)h0);
    float ax1 = sampx((float)(w0 + 15), (float)h0);
    float ax2 = sampx((float)w0,        (float)(h0 + 15));
    float ax3 = sampx((float)(w0 + 15), (float)(h0 + 15));
    float ay0 = sampy((float)w0,        (float)h0);
    float ay1 = sampy((float)(w0 + 15), (float)h0);
    float ay2 = sampy((float)w0,        (float)(h0 + 15));
    float ay3 = sampy((float)(w0 + 15), (float)(h0 + 15));
    float minx = fminf(fminf(ax0, ax1), fminf(ax2, ax3));
    float miny = fminf(fminf(ay0, ay1), fminf(ay2, ay3));

    // Tile origin, clamped so the staged window lies fully inside the image
    // (no reliance on TDM out-of-bounds semantics).
    int x0 = (int)floorf(minx) - 1;
    int y0 = (int)floorf(miny) - 1;
    x0 = min(max(x0, 0), WW - TW);
    y0 = min(max(y0, 0), HH - TH);

    const bool wave0 = (threadIdx.y < 2);  // wave32: lanes 0..31 of the block

    // --- Issue TDM: DMA a 64x40x3 window [x0..x0+63] x [y0..y0+39] x all C
    //     into LDS, packed contiguously [c][y][x] with row pitch TW. ---
    if (wave0) {
        uint32_t lds_off = (uint32_t)(uintptr_t)(void*)sh;
        uint64_t gaddr = (uint64_t)(uintptr_t)(img +
            ((size_t)n * CC * HH * WW + (size_t)y0 * WW + x0));
        // D# group 0: count=1 | lds_addr | global_addr[56:0] | type=2
        u32x4 g0 = { 1u,
                     lds_off,
                     (uint32_t)gaddr,
                     (uint32_t)(((gaddr >> 32) & 0x01FFFFFFu) | 0x80000000u) };
        // D# group 1: data_size=4B; tensor_dim0=W, tensor_dim1=H;
        //             tile = TW x TH x CC; dim0_stride=W; dim1_stride=H*W.
        i32x8 g1 = { (int)(2u << 16),                              // data_size=2 (4B)
                     (int)((uint32_t)WW << 16),                    // tensor_dim0 lo16
                     (int)(((uint32_t)HH & 0xFFFFu) << 16),        // dim0 hi=0 | tensor_dim1 lo16
                     (int)((uint32_t)TW << 16),                    // dim1 hi=0 | tile_dim0
                     (int)((uint32_t)TH | ((uint32_t)CC << 16)),   // tile_dim1 | tile_dim2
                     (int)WW,                                      // tensor_dim0_stride lo32
                     (int)(((uint32_t)(HH * WW) & 0xFFFFu) << 16), // stride0 hi=0 | stride1 lo16
                     (int)((uint32_t)(HH * WW) >> 16) };           // stride1 bits 47:16
        // D# group 2: tensor_dim2 = C; group 3: unused.
        i32x4 g2 = { CC, 0, 0, 0 };
        i32x4 g3 = { 0, 0, 0, 0 };
        // 6-arg toolchain form: trailing i32x8 (unused extension state) = 0.
        i32x8 g4 = { 0, 0, 0, 0, 0, 0, 0, 0 };
        __builtin_amdgcn_tensor_load_to_lds(g0, g1, g2, g3, g4, 0);
    }

    // --- Overlap DMA with per-pixel coordinate / weight math. ---
    const int w = w0 + (int)threadIdx.x;
    const int h = h0 + (int)threadIdx.y;

    float ix = sampx((float)w, (float)h);
    float iy = sampy((float)w, (float)h);
    float fx = floorf(ix), fy = floorf(iy);
    int ix0 = (int)fx, iy0 = (int)fy;
    float tx = ix - fx, ty = iy - fy;

    float wxv[4] = { cub2(tx + 1.0f), cub1(tx), cub1(1.0f - tx), cub2(2.0f - tx) };
    float wyv[4] = { cub2(ty + 1.0f), cub1(ty), cub1(1.0f - ty), cub2(2.0f - ty) };

    // Fast path iff the whole 4x4 tap window sits inside the staged tile.
    // (tile is inside the image, so in-tile => in-image => all weights valid.)
    const bool fast = (ix0 >= x0 + 1) && (ix0 <= x0 + TW - 3) &&
                      (iy0 >= y0 + 1) && (iy0 <= y0 + TH - 3);

    if (wave0) __builtin_amdgcn_s_wait_tensorcnt(0);
    __syncthreads();
    // Prevent the compiler from treating never-VALU-stored LDS as undef
    // (it was written by the TDM, which it cannot see).
    asm volatile("" : : "r"(&sh[0]) : "memory");

    if (w >= WW || h >= HH) return;

    float acc[CC] = {0.0f, 0.0f, 0.0f};

    if (fast) {
        const int lx = ix0 - 1 - x0;
        const int ly = iy0 - 1 - y0;
#pragma unroll
        for (int c = 0; c < CC; ++c) {
            const float* p = &sh[c * (TH * TW) + ly * TW + lx];
            float a = 0.0f;
#pragma unroll
            for (int i = 0; i < 4; ++i) {
                const float* r = p + i * TW;
                float row = fmaf(wxv[0], r[0],
                            fmaf(wxv[1], r[1],
                            fmaf(wxv[2], r[2], wxv[3] * r[3])));
                a = fmaf(wyv[i], row, a);
            }
            acc[c] = a;
        }
    } else {
        // General path: per-axis clamp + weight zeroing (matches reference).
        int xc[4], yc[4];
        float wxm[4], wym[4];
#pragma unroll
        for (int j = 0; j < 4; ++j) {
            int xi = ix0 + j - 1;
            bool okx = (xi >= 0) && (xi < WW);
            xc[j] = okx ? xi : 0;
            wxm[j] = okx ? wxv[j] : 0.0f;
            int yi = iy0 + j - 1;
            bool oky = (yi >= 0) && (yi < HH);
            yc[j] = oky ? yi : 0;
            wym[j] = oky ? wyv[j] : 0.0f;
        }
#pragma unroll
        for (int c = 0; c < CC; ++c) {
            const float* p = img + ((size_t)n * CC + c) * (size_t)(HH * WW);
            float a = 0.0f;
#pragma unroll
            for (int i = 0; i < 4; ++i) {
                const float* r = p + (size_t)yc[i] * WW;
                float row = fmaf(wxm[0], r[xc[0]],
                            fmaf(wxm[1], r[xc[1]],
                            fmaf(wxm[2], r[xc[2]], wxm[3] * r[xc[3]])));
                a = fmaf(wym[i], row, a);
            }
            acc[c] = a;
        }
    }

    const size_t hw = (size_t)HH * WW;
    size_t ob = ((size_t)n * CC) * hw + (size_t)h * WW + w;
#pragma unroll
    for (int c = 0; c < CC; ++c) out[ob + c * hw] = acc[c];
}

extern "C" void kernel_launch(void* const* d_in, const int* in_sizes, int n_in,
                              void* d_out, int out_size, void* d_ws, size_t ws_size,
                              hipStream_t stream) {
    (void)in_sizes; (void)n_in; (void)out_size; (void)d_ws; (void)ws_size;
    const float* img = (const float*)d_in[0];  // arg0_1: [2,3,345,456] f32
    const float* mat = (const float*)d_in[1];  // arg1_1: [2,2,3] f32
    float* out = (float*)d_out;                // [2,3,345,456] f32

    dim3 blk(16, 16, 1);
    dim3 grd((WW + 15) / 16, (HH + 15) / 16, NN);
    grid_sample_bicubic_tdm<<<grd, blk, 0, stream>>>(img, mat, out);
}